// GraphMaxPool_85169201479757
// MI455X (gfx1250) — compile-verified
//
#include <hip/hip_runtime.h>
#include <hip/hip_bf16.h>

typedef __attribute__((ext_vector_type(2))) float v2f;
typedef __attribute__((ext_vector_type(4))) float v4f;
typedef __attribute__((ext_vector_type(8))) float v8f;

// Problem constants (match reference)
#define BB   8
#define NN   16384
#define CC   64
#define KK   16
#define OO   64
#define ROWS (BB * NN)        // 131072
#define MT   128              // rows per block in kernel 1
#define NBLK (ROWS / MT)      // 1024
#define AGG_LD 68             // padded LDS row stride (dwords): 16B-aligned rows,
                              // stride mod 64 banks == 4 -> conflict-free b64 A-frag loads

// ---------------------------------------------------------------------------
// Kernel 1: gather -> max -> relu -> (agg @ W) via WMMA f32 16x16x4,
//           writes raw y to d_out and per-block channel partial sums to ws.
// ---------------------------------------------------------------------------
__global__ __launch_bounds__(256) void k_gather_gemm(
    const float* __restrict__ x, const long long* __restrict__ idx,
    const float* __restrict__ W, float* __restrict__ y,
    float* __restrict__ wsum, float* __restrict__ wsumsq)
{
    __shared__ float sAgg[MT * AGG_LD];   // 34816 B
    __shared__ v2f   sWp[4 * 32 * 16];    // 16 KB: [ct][kp][col] -> (W[2kp][c], W[2kp+1][c])
    __shared__ int   sIdx[MT * KK];       // 8 KB
    __shared__ float sPart[2 * 8 * OO];   // 4 KB  [sum | sumsq] per wave per ch

    const int tid   = threadIdx.x;
    const int block = blockIdx.x;
    const int row0  = block * MT;         // global row base (always within one b)
    const int b     = row0 / NN;
    const int n0    = row0 % NN;
    const float* xb = x + (size_t)b * NN * CC;

    // Stage W into fragment-native pair layout: one contiguous b64 per B frag.
    for (int p = tid; p < 4 * 32 * 16; p += 256) {
        const int ct  = p >> 9;           // p / 512
        const int kp  = (p >> 4) & 31;    // pair index: rows 2kp, 2kp+1
        const int col = p & 15;
        v2f w;
        w.x = W[(2 * kp) * OO + ct * 16 + col];
        w.y = W[(2 * kp + 1) * OO + ct * 16 + col];
        sWp[p] = w;
    }
    // Stage idx (int64 -> int32) into LDS.
    const long long* ip = idx + ((size_t)b * NN + n0) * KK;
    for (int i = tid; i < MT * KK; i += 256) sIdx[i] = (int)ip[i];
    __syncthreads();

    // Gather + max over K neighbors + relu -> agg tile in LDS.
    // 16 lanes x float4 = one 256B coalesced line per neighbor row.
    {
        const int c4 = (tid & 15) * 4;
        const int rg = tid >> 4;          // 0..15
        for (int rr = rg; rr < MT; rr += 16) {
            const int* rip = &sIdx[rr * KK];
            v4f m = {-3.402823466e38f, -3.402823466e38f,
                     -3.402823466e38f, -3.402823466e38f};
#pragma unroll
            for (int k = 0; k < KK; ++k) {
                const int nb = rip[k];
                v4f v = *(const v4f*)(xb + (size_t)nb * CC + c4);
                m.x = fmaxf(m.x, v.x);
                m.y = fmaxf(m.y, v.y);
                m.z = fmaxf(m.z, v.z);
                m.w = fmaxf(m.w, v.w);
            }
            m.x = fmaxf(m.x, 0.0f);
            m.y = fmaxf(m.y, 0.0f);
            m.z = fmaxf(m.z, 0.0f);
            m.w = fmaxf(m.w, 0.0f);
            *(v4f*)&sAgg[rr * AGG_LD + c4] = m;
        }
    }
    __syncthreads();

    // WMMA: each wave computes a 16x64 strip (4 tiles of 16x16), K=64 in 16 steps of 4.
    const int wave  = tid >> 5;           // 0..7
    const int lane  = tid & 31;
    const int l16   = lane & 15;          // A row / B col / D col
    const int khalf = lane >> 4;          // 0 or 1
    const int rbase = wave * 16;

    v8f acc[4] = {};
#pragma unroll
    for (int ks = 0; ks < 16; ++ks) {
        const int k0 = ks * 4 + khalf * 2;
        const v2f a = *(const v2f*)&sAgg[(rbase + l16) * AGG_LD + k0]; // contiguous b64
        const int kp = ks * 2 + khalf;    // pair index within sWp
#pragma unroll
        for (int ct = 0; ct < 4; ++ct) {
            const v2f bf = sWp[(ct * 32 + kp) * 16 + l16];             // contiguous b64
            acc[ct] = __builtin_amdgcn_wmma_f32_16x16x4_f32(
                false, a, false, bf, (short)0, acc[ct], false, false);
        }
    }

    // Store y and fold deterministic per-channel partial sums.
#pragma unroll
    for (int ct = 0; ct < 4; ++ct) {
        float s = 0.0f, q = 0.0f;
#pragma unroll
        for (int v = 0; v < 8; ++v) {
            const float val = acc[ct][v];
            const int   m   = khalf * 8 + v;                    // D row within tile
            y[(size_t)(row0 + rbase + m) * OO + ct * 16 + l16] = val;
            s += val;
            q += val * val;
        }
        // lanes L and L^16 hold the same channel (rows 0-7 vs 8-15)
        s += __shfl_xor(s, 16, 32);
        q += __shfl_xor(q, 16, 32);
        if (khalf == 0) {
            sPart[wave * OO + ct * 16 + l16]          = s;
            sPart[8 * OO + wave * OO + ct * 16 + l16] = q;
        }
    }
    __syncthreads();

    if (tid < OO) {
        float s = 0.0f, q = 0.0f;
#pragma unroll
        for (int w = 0; w < 8; ++w) {
            s += sPart[w * OO + tid];
            q += sPart[8 * OO + w * OO + tid];
        }
        wsum[block * OO + tid]   = s;
        wsumsq[block * OO + tid] = q;
    }
}

// ---------------------------------------------------------------------------
// Kernel 2: reduce block partials -> BN scale/shift per channel.
// ---------------------------------------------------------------------------
__global__ __launch_bounds__(64) void k_stats(
    const float* __restrict__ wsum, const float* __restrict__ wsumsq,
    const float* __restrict__ gamma, const float* __restrict__ beta,
    float* __restrict__ ss)
{
    const int c = threadIdx.x;
    float s = 0.0f, q = 0.0f;
    for (int i = 0; i < NBLK; ++i) {
        s += wsum[i * OO + c];
        q += wsumsq[i * OO + c];
    }
    const float inv  = 1.0f / (float)ROWS;
    const float mean = s * inv;
    const float var  = q * inv - mean * mean;
    const float sc   = gamma[c] * rsqrtf(var + 1e-5f);
    ss[c]      = sc;
    ss[OO + c] = beta[c] - mean * sc;
}

// ---------------------------------------------------------------------------
// Kernel 3: apply BN in place on y (float4 vectorized).
// ---------------------------------------------------------------------------
__global__ __launch_bounds__(256) void k_bn(float* __restrict__ y,
                                            const float* __restrict__ ss)
{
    __shared__ float sc[OO], sh[OO];
    if (threadIdx.x < OO) {
        sc[threadIdx.x] = ss[threadIdx.x];
        sh[threadIdx.x] = ss[OO + threadIdx.x];
    }
    __syncthreads();

    const size_t i = ((size_t)blockIdx.x * 256 + threadIdx.x) * 4;
    const int c0 = (int)(i & 63);         // 4 consecutive channels, no wrap (64%4==0)
    float4 v = *(const float4*)(y + i);
    v.x = v.x * sc[c0 + 0] + sh[c0 + 0];
    v.y = v.y * sc[c0 + 1] + sh[c0 + 1];
    v.z = v.z * sc[c0 + 2] + sh[c0 + 2];
    v.w = v.w * sc[c0 + 3] + sh[c0 + 3];
    *(float4*)(y + i) = v;
}

// ---------------------------------------------------------------------------
extern "C" void kernel_launch(void* const* d_in, const int* in_sizes, int n_in,
                              void* d_out, int out_size, void* d_ws, size_t ws_size,
                              hipStream_t stream)
{
    const float*     x     = (const float*)d_in[0];
    const long long* idx   = (const long long*)d_in[1];   // jnp.int64 per reference
    const float*     W     = (const float*)d_in[2];
    const float*     gamma = (const float*)d_in[3];
    const float*     beta  = (const float*)d_in[4];
    float*           y     = (float*)d_out;

    float* ws     = (float*)d_ws;
    float* wsum   = ws;                       // NBLK*64 floats
    float* wsumsq = ws + NBLK * OO;           // NBLK*64 floats
    float* ss     = ws + 2 * NBLK * OO;       // 128 floats (scale | shift)

    k_gather_gemm<<<NBLK, 256, 0, stream>>>(x, idx, W, y, wsum, wsumsq);
    k_stats<<<1, 64, 0, stream>>>(wsum, wsumsq, gamma, beta, ss);

    const int total_vec4 = (ROWS * OO) / 4;   // 2097152
    k_bn<<<total_vec4 / 256, 256, 0, stream>>>(y, ss);
}